// DPS_11544872092184
// MI455X (gfx1250) — compile-verified
//
#include <hip/hip_runtime.h>
#include <math.h>

// ---------------------------------------------------------------------------
// Problem constants (from reference setup_inputs)
// ---------------------------------------------------------------------------
#define B      8
#define HS     32
#define WS     32
#define D      1024          // HS*WS
#define NNOISE 500
#define K      16
#define C      3
#define HW     1024          // H == W
#define P      64            // patch size
#define SH     32            // H/HS
#define SIGMA  0.05f

typedef __attribute__((ext_vector_type(2))) float v2f;
typedef __attribute__((ext_vector_type(8))) float v8f;

// ---------------------------------------------------------------------------
// Kernel 0: zero the indicator accumulator (workspace is not re-poisoned
// between graph replays, so we must clear it deterministically every launch)
// ---------------------------------------------------------------------------
__global__ void zero_f32(float* __restrict__ p, int n) {
    int i = blockIdx.x * blockDim.x + threadIdx.x;
    if (i < n) p[i] = 0.0f;
}

// ---------------------------------------------------------------------------
// Kernel 1: per-batch min-max normalization of scores -> snorm[B][D]
// ---------------------------------------------------------------------------
__global__ void normalize_scores(const float* __restrict__ scores,
                                 float* __restrict__ snorm) {
    const int b = blockIdx.x;
    const int t = threadIdx.x;          // 256 threads
    __shared__ float smn[256];
    __shared__ float smx[256];
    float mn = 3.0e38f, mx = -3.0e38f;
    for (int i = t; i < D; i += 256) {
        float v = scores[b * D + i];
        mn = fminf(mn, v);
        mx = fmaxf(mx, v);
    }
    smn[t] = mn; smx[t] = mx;
    __syncthreads();
    for (int s = 128; s > 0; s >>= 1) {
        if (t < s) {
            smn[t] = fminf(smn[t], smn[t + s]);
            smx[t] = fmaxf(smx[t], smx[t + s]);
        }
        __syncthreads();
    }
    const float lo  = smn[0];
    const float inv = 1.0f / (smx[0] - lo + 1e-5f);
    for (int i = t; i < D; i += 256)
        snorm[b * D + i] = (scores[b * D + i] - lo) * inv;
}

// ---------------------------------------------------------------------------
// Kernel 2: one wave32 per (b, noise-row): perturb, top-16 (lowest-index
// tie-break like lax.top_k), rank-based scatter == sort(idx)+slot add.
// ---------------------------------------------------------------------------
__global__ __launch_bounds__(32)
void topk_scatter(const float* __restrict__ snorm,
                  const float* __restrict__ noise,
                  float* __restrict__ indicators) {
    const int b = blockIdx.x / NNOISE;
    const int r = blockIdx.x % NNOISE;
    const int lane = threadIdx.x;       // 0..31

    __shared__ float vals[D];
    const float* srow = snorm + b * D;
    const float* nrow = noise + ((size_t)b * NNOISE + r) * D;
    for (int i = lane; i < D; i += 32)
        vals[i] = srow[i] + nrow[i] * SIGMA;
    __syncthreads();

    int idxs[K];
    #pragma unroll 1
    for (int iter = 0; iter < K; ++iter) {
        float best = -3.0e38f;
        int   bi   = 1 << 30;
        for (int i = lane; i < D; i += 32) {
            float v = vals[i];
            if (v > best || (v == best && i < bi)) { best = v; bi = i; }
        }
        // wave32 reduction, lowest index wins ties
        #pragma unroll
        for (int off = 16; off > 0; off >>= 1) {
            float ov = __shfl_xor(best, off);
            int   oi = __shfl_xor(bi, off);
            if (ov > best || (ov == best && oi < bi)) { best = ov; bi = oi; }
        }
        idxs[iter] = bi;
        if (lane == 0) vals[bi] = -3.0e38f;   // remove winner
        __syncthreads();
    }

    // slot = rank of index among the 16 winners (ascending) == jnp.sort(idx)
    if (lane < K) {
        int mine = idxs[lane];
        int rank = 0;
        #pragma unroll
        for (int j = 0; j < K; ++j) rank += (idxs[j] < mine) ? 1 : 0;
        atomicAdd(&indicators[((size_t)b * K + rank) * D + mine],
                  1.0f / (float)NNOISE);
    }
}

// ---------------------------------------------------------------------------
// Kernel 3: patch-gather GEMM with V_WMMA_F32_16X16X4_F32.
//   out[b,j,c,y,x] = sum_cell W[b,j,cell] * Xpad[b,c, y+ih*32-16, x+iw*32-16]
//
// One wave per (b, c, y) output row: 4 N-tiles of 16 pixels sharing one A
// fragment -> 4 back-to-back WMMAs per K-step, 4x A reuse. All B-gather
// bounds checks are branchless (address select + value select) so EXEC stays
// all-1s around every WMMA, as the ISA requires.
// 1536 wave-rows total -> 384 blocks x 4 waves.
// ---------------------------------------------------------------------------
#define NT 4   // N-tiles (of 16 pixels) per wave == one full 64-pixel row

__global__ __launch_bounds__(128)
void wmma_patch_gemm(const float* __restrict__ W,    // [B][K][D]
                     const float* __restrict__ X,    // [B][C][HW][HW]
                     float* __restrict__ out) {      // [B][K][C][P][P]
    const int wave = threadIdx.x >> 5;
    const int lane = threadIdx.x & 31;
    const int row_id = blockIdx.x * 4 + wave;            // 0..1535
    const int b = row_id / (C * P);                      // 192 rows per batch
    const int t = row_id % (C * P);
    const int c = t / P;
    const int y = t % P;                                 // output row 0..63

    const int half = lane >> 4;                          // 0: lanes 0-15, 1: 16-31
    const int n    = lane & 15;                          // N index within tile

    // A rows: lane supplies W[b][m=n][*]  (M striped across lanes 0-15 / 16-31)
    const float* Wrow = W + ((size_t)b * K + n) * D;
    const float* Xbc  = X + ((size_t)b * C + c) * HW * HW;

    v8f acc[NT];
    #pragma unroll
    for (int nt = 0; nt < NT; ++nt) acc[nt] = (v8f){};

    for (int k0 = 0; k0 < D; k0 += 4) {
        const int kA = k0 + 2 * half;                    // lanes<16: k0; else k0+2
        v2f a;
        a.x = Wrow[kA];                                  // VGPR0: K = kA
        a.y = Wrow[kA + 1];                              // VGPR1: K = kA+1

        float bv[NT][2];
        #pragma unroll
        for (int comp = 0; comp < 2; ++comp) {
            const int cell = kA + comp;
            const int ih = cell >> 5;
            const int iw = cell & 31;
            const int row = y + ih * SH - 16;
            const bool rok = (unsigned)row < (unsigned)HW;
            const int rbase = rok ? row * HW : 0;        // safe row base
            const int cbase = n + iw * SH - 16;
            #pragma unroll
            for (int nt = 0; nt < NT; ++nt) {
                const int col = cbase + nt * 16;
                const bool ok = rok & ((unsigned)col < (unsigned)HW);
                const int idx = ok ? (rbase + col) : 0;  // always-valid address
                float v = Xbc[idx];                      // unconditional load
                bv[nt][comp] = ok ? v : 0.0f;            // mask OOB to zero
            }
        }

        #pragma unroll
        for (int nt = 0; nt < NT; ++nt) {
            v2f bm; bm.x = bv[nt][0]; bm.y = bv[nt][1];
            // D = A(16x4,f32) * B(4x16,f32) + C — exact-f32 matrix path
            acc[nt] = __builtin_amdgcn_wmma_f32_16x16x4_f32(
                          false, a, false, bm, (short)0, acc[nt], false, false);
        }
    }

    // C/D layout: VGPR r -> M=r (lanes 0-15) / M=r+8 (lanes 16-31), N=n
    #pragma unroll
    for (int nt = 0; nt < NT; ++nt) {
        const int x = nt * 16 + n;
        #pragma unroll
        for (int rreg = 0; rreg < 8; ++rreg) {
            const int j = rreg + 8 * half;
            out[((((size_t)b * K + j) * C + c) * P + y) * P + x] = acc[nt][rreg];
        }
    }
}

// ---------------------------------------------------------------------------
// Launch
// ---------------------------------------------------------------------------
extern "C" void kernel_launch(void* const* d_in, const int* in_sizes, int n_in,
                              void* d_out, int out_size, void* d_ws, size_t ws_size,
                              hipStream_t stream) {
    const float* scores = (const float*)d_in[0];   // (8,32,32)
    const float* x_high = (const float*)d_in[1];   // (8,3,1024,1024)
    const float* noise  = (const float*)d_in[2];   // (8,500,1024)
    float* out = (float*)d_out;                    // (8,16,3,64,64)

    float* snorm      = (float*)d_ws;              // B*D floats
    float* indicators = snorm + B * D;             // B*K*D floats

    const int n_ind = B * K * D;                   // 131072
    zero_f32<<<(n_ind + 255) / 256, 256, 0, stream>>>(indicators, n_ind);
    normalize_scores<<<B, 256, 0, stream>>>(scores, snorm);
    topk_scatter<<<B * NNOISE, 32, 0, stream>>>(snorm, noise, indicators);

    const int rows = B * C * P;                    // 1536 wave-rows
    wmma_patch_gemm<<<rows / 4, 128, 0, stream>>>(indicators, x_high, out);
}